// ConvAttention_962072674721
// MI455X (gfx1250) — compile-verified
//
#include <hip/hip_runtime.h>
#include <hip/hip_bf16.h>

typedef __attribute__((ext_vector_type(2))) float v2f;
typedef __attribute__((ext_vector_type(8))) float v8f;

#define NTOK 2304   // 48*48
#define CDIM 256
#define HEADS 8
#define DHEAD 32

// ---------------------------------------------------------------------------
// WMMA f32 16x16x4 operand conventions used throughout (wave32):
//   A (16x4, MxK):  lane<16 : vgpr0=A[lane][k+0], vgpr1=A[lane][k+1]
//                   lane>=16: vgpr0=A[lane-16][k+2], vgpr1=A[lane-16][k+3]
//   B (4x16, KxN):  lane<16 : vgpr0=B[k+0][lane], vgpr1=B[k+1][lane]
//                   lane>=16: vgpr0=B[k+2][lane-16], vgpr1=B[k+3][lane-16]
//   C/D (16x16):    vgpr v = C[v + 8*(lane>=16)][lane&15]
// ---------------------------------------------------------------------------

// C[b](MxN) = A(MxK=256) * B[b](256xN) (+bias), all row-major fp32.
// One wave per block; wave computes a 16(M) x 64(N) tile: one A float2 load
// feeds 4 WMMAs per K-step of 4.
__global__ __launch_bounds__(32) void gemm_f32_wmma(
    const float* __restrict__ A, const float* __restrict__ Bm,
    float* __restrict__ Cm, int M, const float* __restrict__ bias) {
  const int lane = threadIdx.x & 31;
  const int lrow = lane & 15;
  const int lh   = lane >> 4;
  const int n0   = blockIdx.x * 64;
  const int m0   = blockIdx.y * 16;
  const int b    = blockIdx.z;

  const float* __restrict__ Bb = Bm + (size_t)b * CDIM * NTOK;
  float* __restrict__       Cb = Cm + (size_t)b * M * NTOK;
  const float* __restrict__ Ar = A + (size_t)(m0 + lrow) * CDIM;

  v8f z = {};
  v8f acc[4] = {z, z, z, z};

#pragma unroll 4
  for (int kk = 0; kk < CDIM / 4; ++kk) {
    const int kc = kk * 4 + 2 * lh;
    v2f a = *(const v2f*)(Ar + kc);                 // A[lrow][kc], A[lrow][kc+1]
    const float* B0 = Bb + (size_t)kc * NTOK + n0 + lrow;
#pragma unroll
    for (int t = 0; t < 4; ++t) {
      v2f bv;
      bv.x = B0[t * 16];                            // B[kc  ][n0+t*16+lrow]
      bv.y = B0[NTOK + t * 16];                     // B[kc+1][n0+t*16+lrow]
      acc[t] = __builtin_amdgcn_wmma_f32_16x16x4_f32(
          false, a, false, bv, (short)0, acc[t], false, false);
    }
  }

  float bias_v[8];
#pragma unroll
  for (int v = 0; v < 8; ++v)
    bias_v[v] = bias ? bias[m0 + v + 8 * lh] : 0.0f;

#pragma unroll
  for (int v = 0; v < 8; ++v) {
    float* cp = Cb + (size_t)(m0 + v + 8 * lh) * NTOK + n0 + lrow;
#pragma unroll
    for (int t = 0; t < 4; ++t) cp[t * 16] = acc[t][v] + bias_v[v];
  }
}

// Flash attention over qkv[b][o][n] (o = {q:0,k:256,v:512} + h*32 + d).
// Block = 4 waves; wave w owns i-tile [i0, i0+16); block covers 64 i rows.
// Writes att[b][h*32+d][n] = softmax(qT k * scale) vT, normalized.
__global__ __launch_bounds__(128) void attn_flash_wmma(
    const float* __restrict__ qkv, float* __restrict__ att) {
  const int tid  = threadIdx.x;
  const int wv   = tid >> 5;
  const int lane = tid & 31;
  const int lrow = lane & 15;
  const int lh   = lane >> 4;
  const int h    = blockIdx.y;
  const int b    = blockIdx.z;
  const int i0   = blockIdx.x * 64 + wv * 16;

  const float* __restrict__ Q  = qkv + ((size_t)b * 768 + h * DHEAD) * NTOK;
  const float* __restrict__ Kp = Q + (size_t)256 * NTOK;
  const float* __restrict__ Vp = Q + (size_t)512 * NTOK;

  __shared__ float v_lds[DHEAD * 16];   // V tile, [d][j], shared by all 4 waves
  __shared__ float p_lds[4][16 * 16];   // per-wave P tile, [i][j]
  __shared__ float r_lds[4][16];        // per-wave row-broadcast (alpha / lsum)

  const float scale = 0.17677669529663687f;  // 32^-0.5

  // Preload Q as A-matrices (A[i][d] = q[d][i]), scale folded in.
  v2f aq[8];
#pragma unroll
  for (int kk = 0; kk < 8; ++kk) {
    const int d0 = 4 * kk + 2 * lh;
    aq[kk].x = Q[(size_t)d0 * NTOK + i0 + lrow] * scale;
    aq[kk].y = Q[(size_t)(d0 + 1) * NTOK + i0 + lrow] * scale;
  }

  float mrow[8], lsum[8];
#pragma unroll
  for (int v = 0; v < 8; ++v) { mrow[v] = -3.0e38f; lsum[v] = 0.0f; }
  v8f acc0 = {};
  v8f acc1 = {};

  for (int jt = 0; jt < NTOK / 16; ++jt) {
    const int j0 = jt * 16;

    __syncthreads();
    // Stage V tile (32 d x 16 j) to LDS with coalesced global loads.
#pragma unroll
    for (int r = 0; r < 4; ++r) {
      const int idx = r * 128 + tid;                 // idx = d*16 + j
      v_lds[idx] = Vp[(size_t)(idx >> 4) * NTOK + j0 + (idx & 15)];
    }
    __syncthreads();

    // S = (Q*scale)^T K : 8 WMMAs over d=32.
    v8f s = {};
#pragma unroll
    for (int kk = 0; kk < 8; ++kk) {
      const int d0 = 4 * kk + 2 * lh;
      v2f bk;
      bk.x = Kp[(size_t)d0 * NTOK + j0 + lrow];
      bk.y = Kp[(size_t)(d0 + 1) * NTOK + j0 + lrow];
      s = __builtin_amdgcn_wmma_f32_16x16x4_f32(
          false, aq[kk], false, bk, (short)0, s, false, false);
    }

    // Online softmax: row (i) reductions live in 16-lane halves.
    float alpha[8];
#pragma unroll
    for (int v = 0; v < 8; ++v) {
      float sv = s[v];
      float tm = sv;
      tm = fmaxf(tm, __shfl_xor(tm, 1, 32));
      tm = fmaxf(tm, __shfl_xor(tm, 2, 32));
      tm = fmaxf(tm, __shfl_xor(tm, 4, 32));
      tm = fmaxf(tm, __shfl_xor(tm, 8, 32));
      const float mn = fmaxf(mrow[v], tm);
      const float al = __expf(mrow[v] - mn);
      const float pv = __expf(sv - mn);
      float ts = pv;
      ts += __shfl_xor(ts, 1, 32);
      ts += __shfl_xor(ts, 2, 32);
      ts += __shfl_xor(ts, 4, 32);
      ts += __shfl_xor(ts, 8, 32);
      lsum[v] = lsum[v] * al + ts;
      mrow[v] = mn;
      alpha[v] = al;
      p_lds[wv][(v + 8 * lh) * 16 + lrow] = pv;      // P[i][j]
    }
    // Broadcast per-row alpha to all lanes via per-wave LDS row.
    if (lrow == 0) {
#pragma unroll
      for (int v = 0; v < 8; ++v) r_lds[wv][v + 8 * lh] = alpha[v];
    }
    const float ai = r_lds[wv][lrow];                // alpha for i = lrow
#pragma unroll
    for (int v = 0; v < 8; ++v) { acc0[v] *= ai; acc1[v] *= ai; }

    // out[d][i] += V * P^T : A[d][j]=v_lds, B[j][i]=P[i][j] from LDS.
    const v2f* p2 = (const v2f*)p_lds[wv];
    const v2f* vt = (const v2f*)v_lds;
#pragma unroll
    for (int kk = 0; kk < 4; ++kk) {
      const int jc = 4 * kk + 2 * lh;
      v2f bp  = p2[(lrow * 16 + jc) >> 1];           // P[lrow][jc], P[lrow][jc+1]
      v2f av0 = vt[(lrow * 16 + jc) >> 1];           // V[d=lrow   ][jc..jc+1]
      v2f av1 = vt[((16 + lrow) * 16 + jc) >> 1];    // V[d=16+lrow][jc..jc+1]
      acc0 = __builtin_amdgcn_wmma_f32_16x16x4_f32(
          false, av0, false, bp, (short)0, acc0, false, false);
      acc1 = __builtin_amdgcn_wmma_f32_16x16x4_f32(
          false, av1, false, bp, (short)0, acc1, false, false);
    }
  }

  // Normalize by row sums and store att[b][h*32+d][i].
  if (lrow == 0) {
#pragma unroll
    for (int v = 0; v < 8; ++v) r_lds[wv][v + 8 * lh] = lsum[v];
  }
  const float rin = 1.0f / r_lds[wv][lrow];
  float* __restrict__ O = att + ((size_t)b * 256 + h * DHEAD) * NTOK;
#pragma unroll
  for (int v = 0; v < 8; ++v) {
    O[(size_t)(v + 8 * lh) * NTOK + i0 + lrow]      = acc0[v] * rin;
    O[(size_t)(16 + v + 8 * lh) * NTOK + i0 + lrow] = acc1[v] * rin;
  }
}

extern "C" void kernel_launch(void* const* d_in, const int* in_sizes, int n_in,
                              void* d_out, int out_size, void* d_ws, size_t ws_size,
                              hipStream_t stream) {
  const float* x     = (const float*)d_in[0];   // (4,256,48,48)
  const float* w_qkv = (const float*)d_in[1];   // (768,256)
  const float* w_out = (const float*)d_in[2];   // (256,256)
  const float* b_out = (const float*)d_in[3];   // (256,)
  float* out = (float*)d_out;                   // (4,256,48,48)

  // Workspace: qkv (4*768*2304 f32 = 28.3MB) then att (4*256*2304 = 9.4MB).
  float* qkv = (float*)d_ws;
  float* att = qkv + (size_t)4 * 768 * NTOK;

  // 1) QKV projection: per batch (768x256)*(256x2304).
  dim3 g1(NTOK / 64, 768 / 16, 4);
  gemm_f32_wmma<<<g1, 32, 0, stream>>>(w_qkv, x, qkv, 768, nullptr);

  // 2) Flash attention per (b,h), 64 query rows per block.
  dim3 g2(NTOK / 64, HEADS, 4);
  attn_flash_wmma<<<g2, 128, 0, stream>>>(qkv, att);

  // 3) Output projection + bias: per batch (256x256)*(256x2304).
  dim3 g3(NTOK / 64, 256 / 16, 4);
  gemm_f32_wmma<<<g3, 32, 0, stream>>>(w_out, att, out, 256, b_out);
}